// HyenaFilter_32796370272335
// MI455X (gfx1250) — compile-verified
//
#include <hip/hip_runtime.h>
#include <hip/hip_bf16.h>
#include <math.h>

// ---------------------------------------------------------------------------
// Problem constants (from setup_inputs): B=4, D=768, L=8192, order=64, emb=33
// ---------------------------------------------------------------------------
#define L_SEQ   8192
#define NFFT    16384        // 2*L = 4^7
#define D_DIM   768
#define B_DIM   4
#define ORDER   64
#define EMB     33
#define KPAD    36           // EMB padded to multiple of 4 for WMMA K-steps
#define NT      512          // threads for FFT kernels (16 waves, 1 WG/WGP)

typedef float v2f __attribute__((ext_vector_type(2)));
typedef float v8f __attribute__((ext_vector_type(8)));

// ===========================================================================
// Kernel 1: sine-MLP producing H (L x 64) with V_WMMA_F32_16X16X4_F32.
// Each wave owns a 16-row tile; 8 waves/WG -> 128 rows/WG; 64 WGs.
// ===========================================================================
__device__ __forceinline__ void gemm16x64(const float (*A)[ORDER],
                                          const float (*W)[ORDER],
                                          int Ksz, int lane, v8f acc[4]) {
  const int m  = lane & 15;
  const int kh = (lane >> 4) << 1;       // 0 or 2
  #pragma unroll
  for (int nt = 0; nt < 4; ++nt) {
    v8f c = {};
    for (int kk = 0; kk < Ksz; kk += 4) {
      v2f a, b;
      a.x = A[m][kk + kh + 0];
      a.y = A[m][kk + kh + 1];
      b.x = W[kk + kh + 0][nt * 16 + m];
      b.y = W[kk + kh + 1][nt * 16 + m];
      c = __builtin_amdgcn_wmma_f32_16x16x4_f32(false, a, false, b,
                                                (short)0, c, false, false);
    }
    acc[nt] = c;
  }
}

__global__ void __launch_bounds__(256)
hyena_mlp_kernel(const float* __restrict__ freq,
                 const float* __restrict__ w0, const float* __restrict__ b0,
                 const float* __restrict__ w1, const float* __restrict__ b1,
                 const float* __restrict__ w2, const float* __restrict__ b2,
                 float* __restrict__ Hout) {
  __shared__ float s_w0[KPAD][ORDER];
  __shared__ float s_w1[ORDER][ORDER];
  __shared__ float s_w2[ORDER][ORDER];
  __shared__ float s_b[3][ORDER];
  __shared__ float s_freq[ORDER];
  __shared__ float s_A[8][16][ORDER];   // per-wave A/h staging

  const int tid = threadIdx.x;
  for (int i = tid; i < KPAD * ORDER; i += 256) {
    int r = i / ORDER, c = i % ORDER;
    s_w0[r][c] = (r < EMB) ? w0[r * ORDER + c] : 0.f;
  }
  for (int i = tid; i < ORDER * ORDER; i += 256) {
    s_w1[i / ORDER][i % ORDER] = w1[i];
    s_w2[i / ORDER][i % ORDER] = w2[i];
  }
  if (tid < ORDER) {
    s_b[0][tid] = b0[tid]; s_b[1][tid] = b1[tid]; s_b[2][tid] = b2[tid];
    s_freq[tid] = freq[tid];
  }
  __syncthreads();

  const int wave = tid >> 5;
  const int lane = tid & 31;
  const int row0 = (blockIdx.x * 8 + wave) * 16;
  float (*A)[ORDER] = s_A[wave];
  const int m  = lane & 15;
  const int kh = lane >> 4;

  // ---- positional embedding z -> A tile (cols >= 33 zeroed) ----------------
  for (int i = lane; i < 16 * ORDER; i += 32) {
    int r = i >> 6, c = i & 63;
    int t = row0 + r;
    float v = 0.f;
    if (c == 0) {
      v = (float)t / (float)(L_SEQ - 1);                    // linspace(0,1,L)
    } else if (c <= 32) {
      int j = (c <= 16) ? (c - 1) : (c - 17);
      float f = 1e-4f + (float)j * (15.0f - 1e-4f) / 15.0f; // linspace(1e-4,15,16)
      float w = 6.28318530717958647f * (float)t / (float)L_SEQ;
      float sv, cv; __sincosf(f * w, &sv, &cv);
      v = (c <= 16) ? cv : (-sv);                           // exp(-i f w): re, im
    }
    A[r][c] = v;
  }
  __builtin_amdgcn_wave_barrier();

  v8f acc[4];
  const float (*Ws[3])[ORDER] = { s_w0, s_w1, s_w2 };
  const int   Ks[3] = { KPAD, ORDER, ORDER };
  for (int layer = 0; layer < 3; ++layer) {
    gemm16x64(A, Ws[layer], Ks[layer], lane, acc);
    __builtin_amdgcn_wave_barrier();
    #pragma unroll
    for (int nt = 0; nt < 4; ++nt)
      #pragma unroll
      for (int i = 0; i < 8; ++i) {
        int M = i + 8 * kh, N = nt * 16 + m;                // C/D layout (ISA 7.12.2)
        A[M][N] = __sinf(s_freq[N] * (acc[nt][i] + s_b[layer][N]));
      }
    __builtin_amdgcn_wave_barrier();
  }
  for (int i = lane; i < 16 * ORDER; i += 32) {
    int r = i >> 6, c = i & 63;
    Hout[(size_t)(row0 + r) * ORDER + c] = A[r][c];
  }
}

// ===========================================================================
// Complex helpers + in-LDS radix-4 FFT (N = 16384 = 4^7).
// fft4_dif : natural in  -> base4-digit-reversed out (forward, e^{-i})
// ifft4_dit: reversed in -> natural out (inverse adjoint, e^{+i}, UNSCALED)
// TW[j] = exp(-2*pi*i*j/NFFT), j in [0, N/2).  W^3 derived as W^1 * W^2.
// ===========================================================================
__device__ __forceinline__ float2 cmul(float2 a, float2 b) {
  return make_float2(a.x * b.x - a.y * b.y, a.x * b.y + a.y * b.x);
}
__device__ __forceinline__ float2 cmulc(float2 a, float2 b) {   // a * conj(b)
  return make_float2(a.x * b.x + a.y * b.y, a.y * b.x - a.x * b.y);
}
__device__ __forceinline__ float2 cadd(float2 a, float2 b) { return make_float2(a.x + b.x, a.y + b.y); }
__device__ __forceinline__ float2 csub(float2 a, float2 b) { return make_float2(a.x - b.x, a.y - b.y); }

__device__ __forceinline__ void fft4_dif(float2* X, const float2* TW, int tid) {
  for (int s = 0; s < 7; ++s) {
    const int q = NFFT >> (2 * s + 2);           // m/4
    __syncthreads();
    for (int idx = tid; idx < NFFT / 4; idx += NT) {
      int blk = idx >> (12 - 2 * s);
      int off = idx & (q - 1);
      int i0  = blk * (q << 2) + off;
      float2 a = X[i0], b = X[i0 + q], c = X[i0 + 2 * q], d = X[i0 + 3 * q];
      float2 t0 = cadd(a, c), t1 = csub(a, c), t2 = cadd(b, d), t3 = csub(b, d);
      float2 r0 = cadd(t0, t2);
      float2 r2 = csub(t0, t2);
      float2 r1 = make_float2(t1.x + t3.y, t1.y - t3.x);    // t1 - i*t3
      float2 r3 = make_float2(t1.x - t3.y, t1.y + t3.x);    // t1 + i*t3
      float2 w1 = TW[off << (2 * s)];
      float2 w2 = TW[(2 * off) << (2 * s)];
      float2 w3 = cmul(w1, w2);
      X[i0]         = r0;
      X[i0 + q]     = cmul(r1, w1);
      X[i0 + 2 * q] = cmul(r2, w2);
      X[i0 + 3 * q] = cmul(r3, w3);
    }
  }
  __syncthreads();
}

__device__ __forceinline__ void ifft4_dit(float2* X, const float2* TW, int tid) {
  for (int s = 6; s >= 0; --s) {
    const int q = NFFT >> (2 * s + 2);
    __syncthreads();
    for (int idx = tid; idx < NFFT / 4; idx += NT) {
      int blk = idx >> (12 - 2 * s);
      int off = idx & (q - 1);
      int i0  = blk * (q << 2) + off;
      float2 w1 = TW[off << (2 * s)];
      float2 w2 = TW[(2 * off) << (2 * s)];
      float2 w3 = cmul(w1, w2);
      float2 a = X[i0];
      float2 b = cmulc(X[i0 + q],     w1);
      float2 c = cmulc(X[i0 + 2 * q], w2);
      float2 d = cmulc(X[i0 + 3 * q], w3);
      float2 u0 = cadd(a, c), u1 = csub(a, c), u2 = cadd(b, d), u3 = csub(b, d);
      X[i0]         = cadd(u0, u2);
      X[i0 + 2 * q] = csub(u0, u2);
      X[i0 + q]     = make_float2(u1.x - u3.y, u1.y + u3.x); // u1 + i*u3
      X[i0 + 3 * q] = make_float2(u1.x + u3.y, u1.y - u3.x); // u1 - i*u3
    }
  }
  __syncthreads();
}

__device__ __forceinline__ void fill_twiddles(float2* TW, int tid) {
  for (int i = tid; i < NFFT / 2; i += NT) {
    float ang = -6.28318530717958647f * (float)i / (float)NFFT;
    float s, c; __sincosf(ang, &s, &c);
    TW[i] = make_float2(c, s);
  }
}

// ===========================================================================
// Kernel 2: per-channel filter row + forward FFT.
// k[d,t] = (H[t,:] . w3[:,d]) * exp(-t/(L-1) * |delta_d|)
// Kf[d,:] = DIF4_FFT(zero-pad(k[d,:])) / NFFT   (digit-reversed order)
// ===========================================================================
__global__ void __launch_bounds__(NT)
hyena_kfft_kernel(const float* __restrict__ H, const float* __restrict__ w3,
                  float2* __restrict__ Kf) {
  __shared__ float2 X[NFFT];          // 128 KB
  __shared__ float2 TW[NFFT / 2];     //  64 KB
  __shared__ float  s_w3[ORDER];
  const int d = blockIdx.x;
  const int tid = threadIdx.x;
  if (tid < ORDER) s_w3[tid] = w3[tid * D_DIM + d];   // w3 is (64, D) row-major
  fill_twiddles(TW, tid);

  const float min_dec = -3.07011345732539450f;  // log(0.01)/1.5
  const float max_dec = -15.3505672866269725f;  // log(0.01)/0.3
  const float delta = fabsf(min_dec + (float)d * (max_dec - min_dec) / (float)(D_DIM - 1));
  __syncthreads();

  for (int t = tid; t < NFFT; t += NT) {
    float v = 0.f;
    if (t < L_SEQ) {
      const float* hrow = H + (size_t)t * ORDER;
      float acc = 0.f;
      #pragma unroll
      for (int k2 = 0; k2 < ORDER; ++k2) acc += hrow[k2] * s_w3[k2];
      float tn = (float)t / (float)(L_SEQ - 1);
      v = acc * __expf(-tn * delta);
    }
    X[t] = make_float2(v, 0.f);
  }
  fft4_dif(X, TW, tid);

  const float sc = 1.0f / (float)NFFT;            // reference: rfft(k)/fft_size
  float2* out = Kf + (size_t)d * NFFT;
  for (int j = tid; j < NFFT; j += NT) {
    float2 v = X[j];
    out[j] = make_float2(v.x * sc, v.y * sc);
  }
}

// ===========================================================================
// Kernel 3: batch-pair packed FFT convolution + bias skip.
// z = x[b0,d,:] + i*x[b1,d,:];  by linearity IDFT(Kf * DFT(z)) = y_b0 + i*y_b1
// (both real), so the pointwise product needs no spectrum unpacking even in
// the digit-reversed domain.  irfft(norm="forward") is unscaled; 1/N is in Kf.
// ===========================================================================
__global__ void __launch_bounds__(NT)
hyena_conv_kernel(const float* __restrict__ x, const float* __restrict__ bias,
                  const float2* __restrict__ Kf, float* __restrict__ y) {
  __shared__ float2 X[NFFT];          // 128 KB
  __shared__ float2 TW[NFFT / 2];     //  64 KB
  const int d   = blockIdx.x % D_DIM;
  const int p   = blockIdx.x / D_DIM;         // batch pair: (2p, 2p+1)
  const int tid = threadIdx.x;
  const float*  x0 = x + ((size_t)(2 * p + 0) * D_DIM + d) * L_SEQ;
  const float*  x1 = x + ((size_t)(2 * p + 1) * D_DIM + d) * L_SEQ;
  float*        y0 = y + ((size_t)(2 * p + 0) * D_DIM + d) * L_SEQ;
  float*        y1 = y + ((size_t)(2 * p + 1) * D_DIM + d) * L_SEQ;
  const float2* kf = Kf + (size_t)d * NFFT;

  fill_twiddles(TW, tid);
  for (int t = tid; t < NFFT; t += NT)
    X[t] = (t < L_SEQ) ? make_float2(x0[t], x1[t]) : make_float2(0.f, 0.f);

  // Prefetch this channel's spectrum (128 KB, one line per 128 B) so it is in
  // L2/WGP$ while the forward FFT runs -> global_prefetch_b8 (CDNA5 path).
  for (int pp = tid; pp < NFFT / 16; pp += NT)
    __builtin_prefetch((const void*)(kf + pp * 16), 0, 1);

  fft4_dif(X, TW, tid);

  for (int j = tid; j < NFFT; j += NT)        // pointwise in scrambled domain
    X[j] = cmul(X[j], kf[j]);
  ifft4_dit(X, TW, tid);

  const float bb = bias[d];
  for (int t = tid; t < L_SEQ; t += NT) {
    float2 v = X[t];
    y0[t] = v.x + x0[t] * bb;
    y1[t] = v.y + x1[t] * bb;
  }
}

// ===========================================================================
// Launch
// ===========================================================================
extern "C" void kernel_launch(void* const* d_in, const int* in_sizes, int n_in,
                              void* d_out, int out_size, void* d_ws, size_t ws_size,
                              hipStream_t stream) {
  (void)in_sizes; (void)n_in; (void)out_size; (void)ws_size;
  const float* x    = (const float*)d_in[0];
  const float* bias = (const float*)d_in[1];
  const float* freq = (const float*)d_in[2];
  const float* w0   = (const float*)d_in[3];
  const float* b0   = (const float*)d_in[4];
  const float* w1   = (const float*)d_in[5];
  const float* b1   = (const float*)d_in[6];
  const float* w2   = (const float*)d_in[7];
  const float* b2   = (const float*)d_in[8];
  const float* w3   = (const float*)d_in[9];
  // d_in[10] is L (int), hardcoded to 8192.

  // Workspace layout: H (L x 64 f32, 2 MB) | Kf (D x NFFT float2, 96 MB)
  float*  H  = (float*)d_ws;
  float2* Kf = (float2*)((char*)d_ws + (size_t)L_SEQ * ORDER * sizeof(float));

  hyena_mlp_kernel <<<L_SEQ / 128,     256, 0, stream>>>(freq, w0, b0, w1, b1, w2, b2, H);
  hyena_kfft_kernel<<<D_DIM,           NT,  0, stream>>>(H, w3, Kf);
  hyena_conv_kernel<<<D_DIM * B_DIM/2, NT,  0, stream>>>(x, bias, Kf, (float*)d_out);
}